// SpatialTransformer_73297911874148
// MI455X (gfx1250) — compile-verified
//
#include <hip/hip_runtime.h>

#define HWD 192
#define PAD 194

typedef float v4f __attribute__((ext_vector_type(4)));

__device__ __forceinline__ int iclamp(int v, int lo, int hi) {
    return v < lo ? lo : (v > hi ? hi : v);
}

// Trilinear sample of the implicitly zero-padded (194^3) volume at continuous
// coords (xs, ys, zs) in padded space; I is the raw 192^3 volume (y,x,z order).
__device__ __forceinline__ float tri_sample(const float* __restrict__ I,
                                            float xs, float ys, float zs) {
    float fx = floorf(xs), fy = floorf(ys), fz = floorf(zs);
    int ix = (int)fx, iy = (int)fy, iz = (int)fz;

    // Padded-space corner indices, clipped to [0, 193] (exactly as reference).
    int x0 = iclamp(ix, 0, PAD - 1);
    int x1 = iclamp(ix + 1, 0, PAD - 1);
    int y0 = iclamp(iy, 0, PAD - 1);
    int y1 = iclamp(iy + 1, 0, PAD - 1);
    int z0 = iclamp(iz, 0, PAD - 1);
    int z1 = iclamp(iz + 1, 0, PAD - 1);

    // Fractional weights from the *clipped* upper corner (reference quirk).
    float ddx = (float)x1 - xs;
    float ddy = (float)y1 - ys;
    float ddz = (float)z1 - zs;
    float ex = 1.0f - ddx, ey = 1.0f - ddy, ez = 1.0f - ddz;

    // Map padded -> unpadded; pad cells (coord 0 or 193) read zero.
    int ux0 = x0 - 1, ux1 = x1 - 1;
    int uy0 = y0 - 1, uy1 = y1 - 1;
    int uz0 = z0 - 1, uz1 = z1 - 1;
    bool mx0 = (unsigned)ux0 < (unsigned)HWD;
    bool mx1 = (unsigned)ux1 < (unsigned)HWD;
    bool my0 = (unsigned)uy0 < (unsigned)HWD;
    bool my1 = (unsigned)uy1 < (unsigned)HWD;
    bool mz0 = (unsigned)uz0 < (unsigned)HWD;
    bool mz1 = (unsigned)uz1 < (unsigned)HWD;

    int cx0 = iclamp(ux0, 0, HWD - 1), cx1 = iclamp(ux1, 0, HWD - 1);
    int cy0 = iclamp(uy0, 0, HWD - 1), cy1 = iclamp(uy1, 0, HWD - 1);
    int cz0 = iclamp(uz0, 0, HWD - 1), cz1 = iclamp(uz1, 0, HWD - 1);

    // Row-plane base offsets (y,x) * 192; strides: y:192*192, x:192, z:1.
    int o00 = (cy0 * HWD + cx0) * HWD;  // (y0, x0)
    int o10 = (cy1 * HWD + cx0) * HWD;  // (y1, x0)
    int o01 = (cy0 * HWD + cx1) * HWD;  // (y0, x1)
    int o11 = (cy1 * HWD + cx1) * HWD;  // (y1, x1)

    float Ia  = (my0 & mx0 & mz0) ? I[o00 + cz0] : 0.0f;
    float Ib  = (my1 & mx0 & mz0) ? I[o10 + cz0] : 0.0f;
    float Ic  = (my0 & mx1 & mz0) ? I[o01 + cz0] : 0.0f;
    float Id  = (my1 & mx1 & mz0) ? I[o11 + cz0] : 0.0f;
    float Ie  = (my0 & mx0 & mz1) ? I[o00 + cz1] : 0.0f;
    float If_ = (my1 & mx0 & mz1) ? I[o10 + cz1] : 0.0f;
    float Ig  = (my0 & mx1 & mz1) ? I[o01 + cz1] : 0.0f;
    float Ih  = (my1 & mx1 & mz1) ? I[o11 + cz1] : 0.0f;

    float wa = ddz * ddx * ddy;
    float wb = ddz * ddx * ey;
    float wc = ddz * ex * ddy;
    float wd = ddz * ex * ey;
    float we = ez * ddx * ddy;
    float wf = ez * ddx * ey;
    float wg = ez * ex * ddy;
    float wh = ez * ex * ey;

    float acc = wa * Ia;
    acc = fmaf(wb, Ib, acc);
    acc = fmaf(wc, Ic, acc);
    acc = fmaf(wd, Id, acc);
    acc = fmaf(we, Ie, acc);
    acc = fmaf(wf, If_, acc);
    acc = fmaf(wg, Ig, acc);
    acc = fmaf(wh, Ih, acc);
    return acc;
}

// Grid: (1, 48, 192), block: (48, 4).
//   h = blockIdx.z, w = 4*blockIdx.y + threadIdx.y, z = 4*threadIdx.x .. +3
// Each thread: 1x b128 NT load per field, 32 L2 gathers, 1x b128 NT store.
// Each 192-thread block touches one contiguous 3KB chunk of each stream.
__global__ __launch_bounds__(192) void SpatialTransformer_73297911874148_kernel(
    const float* __restrict__ I, const float* __restrict__ DX,
    const float* __restrict__ DY, const float* __restrict__ DZ,
    float* __restrict__ O) {
    int d = (int)threadIdx.x * 4;                      // z of first element
    int w = (int)blockIdx.y * 4 + (int)threadIdx.y;    // x index
    int h = (int)blockIdx.z;                           // y index
    size_t v = ((size_t)(h * HWD + w)) * HWD + d;      // flat voxel index

    // Prefetch streaming displacement data ~16KB ahead.
    __builtin_prefetch(DX + v + 4096, 0, 0);
    __builtin_prefetch(DY + v + 4096, 0, 0);
    __builtin_prefetch(DZ + v + 4096, 0, 0);

    // Non-temporal coalesced b128 loads of streaming fields (keep L2 for I).
    v4f dxv = __builtin_nontemporal_load((const v4f*)(DX + v));
    v4f dyv = __builtin_nontemporal_load((const v4f*)(DY + v));
    v4f dzv = __builtin_nontemporal_load((const v4f*)(DZ + v));

    float xb = (float)w + 1.0f;
    float yb = (float)h + 1.0f;
    float zb = (float)d + 1.0f;

    v4f out;
    out.x = tri_sample(I, dxv.x + xb, dyv.x + yb, dzv.x + zb);
    out.y = tri_sample(I, dxv.y + xb, dyv.y + yb, dzv.y + zb + 1.0f);
    out.z = tri_sample(I, dxv.z + xb, dyv.z + yb, dzv.z + zb + 2.0f);
    out.w = tri_sample(I, dxv.w + xb, dyv.w + yb, dzv.w + zb + 3.0f);

    __builtin_nontemporal_store(out, (v4f*)(O + v));
}

extern "C" void kernel_launch(void* const* d_in, const int* in_sizes, int n_in,
                              void* d_out, int out_size, void* d_ws, size_t ws_size,
                              hipStream_t stream) {
    (void)in_sizes; (void)n_in; (void)out_size; (void)d_ws; (void)ws_size;
    const float* I  = (const float*)d_in[0];
    const float* DX = (const float*)d_in[1];
    const float* DY = (const float*)d_in[2];
    const float* DZ = (const float*)d_in[3];
    float* O = (float*)d_out;

    dim3 block(48, 4, 1);      // 192 threads = 6 wave32
    dim3 grid(1, HWD / 4, HWD);
    SpatialTransformer_73297911874148_kernel<<<grid, block, 0, stream>>>(I, DX, DY, DZ, O);
}